// myRNN_85134841741474
// MI455X (gfx1250) — compile-verified
//
#include <hip/hip_runtime.h>

// y[b,s,f] = init[b] + FACTOR * cumsum_s(x[b,:,f])   (B=1024, S=4096, F=11, fp32)
// Bandwidth-bound: ~369 MB traffic -> ~16 us at 23.3 TB/s. Strategy:
//   * 1 workgroup per batch, 352 threads = 11 wave32s (one wave per feature)
//   * double-buffered LDS tiles (512 timesteps), filled with gfx1250 async
//     global->LDS loads (ASYNCcnt), scanned in LDS, streamed out with NT stores
//   * per-wave: chunk-per-lane local sums + 5-step wave32 shuffle scan

typedef float v2f __attribute__((ext_vector_type(2)));
typedef int   v2i __attribute__((ext_vector_type(2)));
typedef v2i __attribute__((address_space(1))) as1_v2i;   // global
typedef v2i __attribute__((address_space(3))) as3_v2i;   // LDS

constexpr int S         = 4096;
constexpr int F         = 11;
constexpr int TILE_S    = 512;                 // timesteps per LDS tile
constexpr int NTILES    = S / TILE_S;          // 8
constexpr int CHUNK_S   = TILE_S / 32;         // 16 timesteps per lane
constexpr int CHUNK_F   = CHUNK_S * F;         // 176 floats per chunk
constexpr int CHUNK_PAD = CHUNK_F + 2;         // pad: stride 178 = 50 mod 64 -> conflict-free
constexpr int BUF_F     = 32 * CHUNK_PAD;      // 5696 floats per buffer (22784 B)
constexpr int TILE_ELEMS= TILE_S * F;          // 5632 floats per tile
constexpr int NTHREADS  = 352;                 // 11 waves
constexpr int LPT       = (TILE_ELEMS / 2) / NTHREADS;  // 8 float2 per thread per tile

#define RNN_FACTOR (-1.0f)

#if defined(__HIP_DEVICE_COMPILE__)
# if __has_builtin(__builtin_amdgcn_global_load_async_to_lds_b64)
#   define HAVE_ASYNC_LDS 1
# else
#   define HAVE_ASYNC_LDS 0
# endif
# if __has_builtin(__builtin_amdgcn_s_wait_asynccnt)
#   define WAIT_ASYNC(n) __builtin_amdgcn_s_wait_asynccnt(n)
# else
#   define WAIT_ASYNC(n) asm volatile("s_wait_asynccnt %0" :: "i"(n) : "memory")
# endif
#else
# define HAVE_ASYNC_LDS 0
# define WAIT_ASYNC(n)
#endif

// Issue one tile's worth of global->LDS copies (8 x float2 per thread).
// Chunk-padded LDS layout: element (s,f) of tile -> (s/16)*178 + (s%16)*11 + f.
// A float2 group never straddles a chunk (176 divisible by 2) and stays 8B aligned.
__device__ __forceinline__ void issue_tile_loads(const float* __restrict__ xt,
                                                 float* __restrict__ buf,
                                                 int tid) {
#pragma unroll
  for (int k = 0; k < LPT; ++k) {
    int g      = tid + k * NTHREADS;        // float2 group index, 0..2815
    int chunk  = g / (CHUNK_F / 2);         // /88
    int within = 2 * g - chunk * CHUNK_F;
    const float* gp = xt + 2 * g;
    float*       lp = buf + chunk * CHUNK_PAD + within;
#if HAVE_ASYNC_LDS
    __builtin_amdgcn_global_load_async_to_lds_b64(
        (as1_v2i*)gp, (as3_v2i*)lp, 0, 0);
#else
    *(v2f*)lp = __builtin_nontemporal_load((const v2f*)gp);
#endif
  }
}

__global__ __launch_bounds__(NTHREADS) void rnn_cumsum_scan(
    const float* __restrict__ x, const float* __restrict__ init0,
    float* __restrict__ y) {
  __shared__ float lds[2 * BUF_F];          // 45568 B double buffer

  const int b   = blockIdx.x;
  const int tid = threadIdx.x;
  const float* xb = x + (size_t)b * (S * F);
  float*       yb = y + (size_t)b * (S * F);

  // Prime the pipeline: tiles 0 and 1 in flight.
  issue_tile_loads(xb + 0 * TILE_ELEMS, lds + 0 * BUF_F, tid);
  issue_tile_loads(xb + 1 * TILE_ELEMS, lds + 1 * BUF_F, tid);

  const int f = tid >> 5;   // wave index == feature (exactly 11 waves)
  const int l = tid & 31;   // lane == chunk index within tile
  float carry = init0[b];   // running state at start of current tile (per wave f)

  for (int t = 0; t < NTILES; ++t) {
    float* buf = lds + (t & 1) * BUF_F;

    // Async loads complete in order: waiting <= LPT means the older tile's
    // LPT loads are done while the newer tile's LPT may still be in flight.
    if (t == NTILES - 1) { WAIT_ASYNC(0); } else { WAIT_ASYNC(LPT); }
    __syncthreads();

    // Pass 1: per-lane chunk total (16 timesteps, stride 11 floats; banks
    // (50*l + 11*j + f) mod 64 are distinct across lanes -> conflict-free).
    float tot = 0.0f;
    {
      int idx = l * CHUNK_PAD + f;
#pragma unroll
      for (int j = 0; j < CHUNK_S; ++j) { tot += buf[idx]; idx += F; }
    }
    // Wave32 inclusive scan of chunk totals.
    float incl = tot;
#pragma unroll
    for (int d = 1; d < 32; d <<= 1) {
      float nb = __shfl_up(incl, d, 32);
      incl += (l >= d) ? nb : 0.0f;
    }
    // Exclusive prefix -> running state entering this lane's chunk.
    float run = fmaf(RNN_FACTOR, incl - tot, carry);
    // Pass 2: apply scan, write results back into LDS in place.
    {
      int idx = l * CHUNK_PAD + f;
#pragma unroll
      for (int j = 0; j < CHUNK_S; ++j) {
        run = fmaf(RNN_FACTOR, buf[idx], run);
        buf[idx] = run;
        idx += F;
      }
    }
    carry = __shfl(run, 31, 32);  // lane 31's final run == state at tile end

    __syncthreads();

    // Coalesced streaming copy-out (non-temporal: no reuse of y).
    float* yt = yb + t * TILE_ELEMS;
#pragma unroll
    for (int k = 0; k < LPT; ++k) {
      int g      = tid + k * NTHREADS;
      int chunk  = g / (CHUNK_F / 2);
      int within = 2 * g - chunk * CHUNK_F;
      v2f v = *(const v2f*)(buf + chunk * CHUNK_PAD + within);
      __builtin_nontemporal_store(v, (v2f*)(yt + 2 * g));
    }

    // Refill this buffer with tile t+2 once everyone is done reading it.
    if (t + 2 < NTILES) {
      __syncthreads();
      issue_tile_loads(xb + (t + 2) * TILE_ELEMS, buf, tid);
    }
  }
}

extern "C" void kernel_launch(void* const* d_in, const int* in_sizes, int n_in,
                              void* d_out, int out_size, void* d_ws, size_t ws_size,
                              hipStream_t stream) {
  (void)n_in; (void)out_size; (void)d_ws; (void)ws_size;
  const float* x    = (const float*)d_in[0];   // [B, S, F] fp32
  const float* init = (const float*)d_in[1];   // [B, 1] fp32
  float* y          = (float*)d_out;           // [B, S, F] fp32
  const int B = in_sizes[1];                   // 1024
  rnn_cumsum_scan<<<dim3(B), dim3(NTHREADS), 0, stream>>>(x, init, y);
}